// CTRNN_8306466750942
// MI455X (gfx1250) — compile-verified
//
#include <hip/hip_runtime.h>
#include <cstddef>
#include <cstdint>

// CTRNN persistent-scan kernel for MI455X (gfx1250, wave32, WMMA).
//
// B=64, T=1024, I=128, H=512, O=64.
// Grid: 16 workgroups x 256 threads (8 wave32) -- persistent/cooperative.
// Wave (wg g, wave w) owns the 16x16 state tile (mt = w&3, nt = 2g + (w>>2));
// h tile lives in 8 accumulator VGPRs for the whole scan (fp32 WMMA C/D layout).
// tanh(h) is broadcast through a double-buffered global array, staged into LDS
// each step with CDNA5 GLOBAL_LOAD_ASYNC_TO_LDS_B128 (ASYNCcnt-tracked).
// Readout for step t is deferred to iteration t+1 so it reads the staged LDS
// copy instead of uncoalesced global loads.
//
// Per-WG LDS (dynamic, 248KB of the 320KB/WGP on CDNA5):
//   A_s    : 64x512 staged tanh(h_prev)         (128KB)
//   Wrec_s : 32x512 slice of W_rec              ( 64KB)
//   Win_s  : 32x128 slice of W_in               ( 16KB)
//   Wout_s : 16x512 slice of W_out              ( 32KB)
//   Red_s  : 8x256 readout partials             (  8KB)

#define B_ 64
#define T_ 1024
#define I_ 128
#define H_ 512
#define O_ 64
#define G_ 16
#define NTHREADS 256
#define DT_ 0.1f

#define USE_ASYNC_LDS 1

typedef float v2f __attribute__((ext_vector_type(2)));
typedef float v4f __attribute__((ext_vector_type(4)));
typedef float v8f __attribute__((ext_vector_type(8)));

#if __has_builtin(__builtin_amdgcn_tanh_f32)
__device__ __forceinline__ float fast_tanh(float x) {
  return __builtin_amdgcn_tanh_f32(x);   // CDNA5 TRANS32 v_tanh_f32
}
#else
__device__ __forceinline__ float fast_tanh(float x) { return tanhf(x); }
#endif

__device__ __forceinline__ v8f wmma_f32_k4(v2f a, v2f b, v8f c) {
  // D = A(16x4 f32) * B(4x16 f32) + C(16x16 f32)
  return __builtin_amdgcn_wmma_f32_16x16x4_f32(
      false, a, false, b, (short)0, c, false, false);
}

// Device-scope grid barrier: monotonic counter (reset by host memset each
// launch), release on arrive, acquire after spin. All 16 WGs participate.
__device__ __forceinline__ void grid_barrier(unsigned* cnt, unsigned idx) {
  __threadfence();            // publish this thread's global stores
  __syncthreads();
  if (threadIdx.x == 0) {
    atomicAdd(cnt, 1u);
    const unsigned target = idx * (unsigned)G_;
    while (__hip_atomic_load(cnt, __ATOMIC_ACQUIRE,
                             __HIP_MEMORY_SCOPE_AGENT) < target) {
      __builtin_amdgcn_s_sleep(8);
    }
  }
  __syncthreads();
  __threadfence();            // acquire: invalidate stale near caches
}

__global__ void __launch_bounds__(NTHREADS, 1)
ctrnn_scan(const float* __restrict__ X,     // (B,T,I)
           const float* __restrict__ H0,    // (B,H)
           const float* __restrict__ Win,   // (H,I)
           const float* __restrict__ bin,   // (H)
           const float* __restrict__ Wrec,  // (H,H)
           const float* __restrict__ Wout,  // (O,H)
           const float* __restrict__ bout,  // (O)
           float* __restrict__ Out,         // (B,T,O) ++ final_h (B,H)
           float* __restrict__ Abuf0,       // (B,H) tanh(h) double buffer 0
           float* __restrict__ Abuf1,       // (B,H) tanh(h) double buffer 1
           unsigned* __restrict__ bar_cnt) {
  extern __shared__ float smem[];
  float* A_s    = smem;                  // 64*512
  float* Wrec_s = A_s + B_ * H_;         // 32*512
  float* Win_s  = Wrec_s + 32 * H_;      // 32*128
  float* Wout_s = Win_s + 32 * I_;       // 16*512
  float* Red_s  = Wout_s + 16 * H_;      // 8*256

  const int tid  = threadIdx.x;
  const int g    = blockIdx.x;
  const int wv   = tid >> 5;             // wave id 0..7 (wave32)
  const int lane = tid & 31;
  const int l16  = lane & 15;
  const int hf   = lane >> 4;            // half-wave select
  const int kh   = hf << 1;              // K sub-offset: 0 or 2

  const int mt  = wv & 3;                // M tile 0..3 (batch rows)
  const int ntl = wv >> 2;               // local N tile 0..1
  const int b0  = mt << 4;
  const int n0  = ((g << 1) + ntl) << 4; // global H column base
  const int n   = n0 + l16;              // this lane's H column

  // ---- preload resident weight slices into LDS (coalesced b128) ----
  {
    const v4f* s = (const v4f*)(Wrec + (size_t)(32 * g) * H_);
    v4f* d = (v4f*)Wrec_s;
    for (int i = tid; i < 32 * H_ / 4; i += NTHREADS) d[i] = s[i];
  }
  {
    const v4f* s = (const v4f*)(Win + (size_t)(32 * g) * I_);
    v4f* d = (v4f*)Win_s;
    for (int i = tid; i < 32 * I_ / 4; i += NTHREADS) d[i] = s[i];
  }
  {
    const v4f* s = (const v4f*)(Wout + (size_t)(16 * (g >> 2)) * H_);
    v4f* d = (v4f*)Wout_s;
    for (int i = tid; i < 16 * H_ / 4; i += NTHREADS) d[i] = s[i];
  }

  // ---- load initial hidden state tile into registers; publish tanh(h0) ----
  float h[8];
#pragma unroll
  for (int v = 0; v < 8; ++v) {
    const int brow = b0 + v + 8 * hf;    // C/D layout: m = v + 8*(lane>=16)
    h[v] = H0[(size_t)brow * H_ + n];
    Abuf0[(size_t)brow * H_ + n] = fast_tanh(h[v]);
  }
  const float bias_n = bin[n];

  // ---- loop-invariant readout mapping (this WG owns out tile g&3, g>>2) ----
  const int bo0 = (g & 3) << 4;
  const int o0  = (g >> 2) << 4;
  const int k0w = wv << 6;               // 64-wide K chunk per wave
  const int rm  = (tid >> 5) + 8 * ((tid & 31) >> 4);  // C/D row of thread
  const int ro  = tid & 15;                            // C/D col of thread
  const float bout_v = bout[o0 + ro];
  float* outp = Out + ((size_t)(bo0 + rm) * T_) * O_ + (o0 + ro);

  unsigned barIdx = 0;
  grid_barrier(bar_cnt, ++barIdx);       // A0 = tanh(h0) visible everywhere

  // Iteration t stages tanh(h entering step t); the readout for step t-1 is
  // done here (it needs tanh(h after step t-1) == staged buffer). One extra
  // iteration (t == T_) performs the final readout only.
  for (int t = 0; t <= T_; ++t) {
    const float* Aprev = (t & 1) ? Abuf1 : Abuf0;
    float*       Anext = (t & 1) ? Abuf0 : Abuf1;

    // ---- stage tanh(h_prev) into LDS ----
#if USE_ASYNC_LDS
    for (int i = tid; i < B_ * H_ / 4; i += NTHREADS) {
      const v4f* src = (const v4f*)Aprev + i;
      unsigned dst = (unsigned)(uintptr_t)(A_s + 4 * (size_t)i);
      asm volatile("global_load_async_to_lds_b128 %0, %1, off"
                   :: "v"(dst), "v"(src)
                   : "memory");
    }
    asm volatile("s_wait_asynccnt 0x0" ::: "memory");
#else
    {
      const v4f* s = (const v4f*)Aprev;
      v4f* d = (v4f*)A_s;
      for (int i = tid; i < B_ * H_ / 4; i += NTHREADS) d[i] = s[i];
    }
#endif
    __syncthreads();

    // ---- deferred readout for step t-1 (reads staged LDS copy) ----
    if (t > 0) {
      v8f co = {};
      const float* arow = A_s + (bo0 + l16) * H_;
      const float* wrow = Wout_s + l16 * H_;
#pragma unroll 4
      for (int k = 0; k < 64; k += 4) {
        v2f a = *(const v2f*)(arow + k0w + k + kh);
        v2f b = *(const v2f*)(wrow + k0w + k + kh);
        co = wmma_f32_k4(a, b, co);
      }
#pragma unroll
      for (int v = 0; v < 8; ++v)
        Red_s[(wv << 8) + (v << 5) + lane] = co[v];
      __syncthreads();
      float s = 0.f;
#pragma unroll
      for (int w2 = 0; w2 < 8; ++w2) s += Red_s[(w2 << 8) + tid];
      outp[(size_t)(t - 1) * O_] = s + bout_v;
      __syncthreads();                   // Red_s safe for next iteration
    }
    if (t == T_) break;

    v8f c = {};

    // ---- input drive: X[:,t,:] (16xI) x Win_sliceT, K=128 => 32 WMMA ----
    {
      const float* xrow = X + ((size_t)(b0 + l16) * T_ + t) * I_;
      __builtin_prefetch(xrow + I_, 0, 0);   // next timestep's row -> L2
      const float* wrow = Win_s + (ntl * 16 + l16) * I_;
#pragma unroll 8
      for (int k = 0; k < I_; k += 4) {
        v2f a = *(const v2f*)(xrow + k + kh);
        v2f b = *(const v2f*)(wrow + k + kh);
        c = wmma_f32_k4(a, b, c);
      }
    }

    // ---- recurrence: tanh(h_prev) (16xH) x Wrec_sliceT, K=512 => 128 WMMA
    {
      const float* arow = A_s + (b0 + l16) * H_;
      const float* wrow = Wrec_s + (ntl * 16 + l16) * H_;
#pragma unroll 8
      for (int k = 0; k < H_; k += 4) {
        v2f a = *(const v2f*)(arow + k + kh);
        v2f b = *(const v2f*)(wrow + k + kh);
        c = wmma_f32_k4(a, b, c);
      }
    }

    // ---- Euler update in registers; publish tanh(h_new) ----
#pragma unroll
    for (int v = 0; v < 8; ++v) {
      h[v] = h[v] + DT_ * (-h[v] + c[v] + bias_n);   // TAU = 1
      const int brow = b0 + v + 8 * hf;
      Anext[(size_t)brow * H_ + n] = fast_tanh(h[v]);
    }

    grid_barrier(bar_cnt, ++barIdx);     // tanh(h_t) complete device-wide
  }

  // ---- final hidden state appended after outputs ----
  float* Hf = Out + (size_t)B_ * T_ * O_;
#pragma unroll
  for (int v = 0; v < 8; ++v) {
    const int brow = b0 + v + 8 * hf;
    Hf[(size_t)brow * H_ + n] = h[v];
  }
}

extern "C" void kernel_launch(void* const* d_in, const int* in_sizes, int n_in,
                              void* d_out, int out_size, void* d_ws,
                              size_t ws_size, hipStream_t stream) {
  const float* X    = (const float*)d_in[0];  // inputs (B,T,I)
  const float* H0   = (const float*)d_in[1];  // hidden_state (B,H)
  const float* Win  = (const float*)d_in[2];  // W_in_w (H,I)
  const float* bin  = (const float*)d_in[3];  // b_in (H)
  const float* Wrec = (const float*)d_in[4];  // W_rec_w (H,H)
  const float* Wout = (const float*)d_in[5];  // W_out_w (O,H)
  const float* bout = (const float*)d_in[6];  // b_out (O)
  float* Out = (float*)d_out;                 // (B,T,O) ++ (B,H)

  unsigned char* ws = (unsigned char*)d_ws;
  unsigned* cnt = (unsigned*)ws;                                    // barrier
  float* A0 = (float*)(ws + 256);                                   // 128KB
  float* A1 = (float*)(ws + 256 + (size_t)B_ * H_ * sizeof(float)); // 128KB

  // Reset barrier counter every launch (graph-capture safe, deterministic).
  (void)hipMemsetAsync(d_ws, 0, 256, stream);

  const size_t SHMEM =
      (size_t)(B_ * H_ + 32 * H_ + 32 * I_ + 16 * H_ + 8 * 256) * sizeof(float);
  (void)hipFuncSetAttribute(reinterpret_cast<const void*>(ctrnn_scan),
                            hipFuncAttributeMaxDynamicSharedMemorySize,
                            (int)SHMEM);

  ctrnn_scan<<<dim3(G_), dim3(NTHREADS), SHMEM, stream>>>(
      X, H0, Win, bin, Wrec, Wout, bout, Out, A0, A1, cnt);
}